// Uni_Conv2D_90855738180336
// MI455X (gfx1250) — compile-verified
//
#include <hip/hip_runtime.h>

// ---------------- problem constants ----------------
#define BATCH   32
#define CIN     64
#define HW      56
#define COUT    128
#define CONN    3
#define EPS     0.01f

// ---------------- tiling ----------------
#define TH      8
#define TW      8
#define TH2     (TH + 2)           // 10 (halo for 3x3, edge-clamped)
#define TW2     (TW + 2)           // 10
#define TILES_X (HW / TW)          // 7
#define TILES_Y (HW / TH)          // 7
#define TILE_DW (CIN * TH2 * TW2)  // 6400 dwords of staged x
#define PARM_DW (COUT * 8)         // 1024 dwords of per-channel params
#define SMEM_DW (TILE_DW + PARM_DW)

typedef __attribute__((address_space(1))) int gas_int;
typedef __attribute__((address_space(3))) int las_int;

// Async gather: per-lane global address -> per-lane LDS address (ASYNCcnt path).
__device__ __forceinline__ void async_g2l_b32(const void* g, void* l) {
#if __has_builtin(__builtin_amdgcn_global_load_async_to_lds_b32)
    __builtin_amdgcn_global_load_async_to_lds_b32((gas_int*)(void*)g,
                                                  (las_int*)l, 0, 0);
#else
    unsigned int loff = (unsigned int)(unsigned long long)(las_int*)l;
    asm volatile("global_load_async_to_lds_b32 %0, %1, off"
                 :: "v"(loff), "v"(g) : "memory");
#endif
}

__device__ __forceinline__ void wait_async_zero() {
#if __has_builtin(__builtin_amdgcn_s_wait_asynccnt)
    __builtin_amdgcn_s_wait_asynccnt(0);
#else
    asm volatile("s_wait_asynccnt 0" ::: "memory");
#endif
}

__global__ __launch_bounds__(256)
void uniconv_gather_kernel(const float* __restrict__ x,
                           const float* __restrict__ weights,
                           const float* __restrict__ bias,
                           const int*   __restrict__ conn,
                           float*       __restrict__ out) {
    __shared__ unsigned int smem[SMEM_DW];

    const int t    = threadIdx.x;
    const int blk  = blockIdx.x;
    const int tile = blk % (TILES_X * TILES_Y);
    const int b    = blk / (TILES_X * TILES_Y);
    const int ty0  = (tile / TILES_X) * TH;
    const int tx0  = (tile % TILES_X) * TW;

    // ---- phase 1a: async-stage edge-clamped input tile (all 64 ch) into LDS ----
    const float* xb = x + (size_t)b * CIN * HW * HW;
    for (int e = t; e < TILE_DW; e += 256) {          // 25 async ops per lane
        int c  = e / (TH2 * TW2);
        int r  = e % (TH2 * TW2);
        int hh = r / TW2;
        int ww = r % TW2;
        int gh = ty0 - 1 + hh;  gh = gh < 0 ? 0 : (gh > HW - 1 ? HW - 1 : gh);
        int gw = tx0 - 1 + ww;  gw = gw < 0 ? 0 : (gw > HW - 1 ? HW - 1 : gw);
        const float* gp = xb + ((size_t)c * HW + gh) * HW + gw;
        async_g2l_b32(gp, &smem[e]);
    }

    // ---- phase 1b: per-channel params (normalized weights + LDS tap offsets) ----
    if (t < COUT) {
        float w0 = weights[t * 3 + 0];
        float w1 = weights[t * 3 + 1];
        float w2 = weights[t * 3 + 2];
        int   a0 = conn[t * 3 + 0];
        int   a1 = conn[t * 3 + 1];
        int   a2 = conn[t * 3 + 2];
        // sumw over the scattered one-hot matrix: merge colliding indices BEFORE abs
        float s;
        if (a0 == a1 && a0 == a2)      s = fabsf(w0 + w1 + w2);
        else if (a0 == a1)             s = fabsf(w0 + w1) + fabsf(w2);
        else if (a0 == a2)             s = fabsf(w0 + w2) + fabsf(w1);
        else if (a1 == a2)             s = fabsf(w1 + w2) + fabsf(w0);
        else                           s = fabsf(w0) + fabsf(w1) + fabsf(w2);
        s += EPS;

        unsigned int* pp = &smem[TILE_DW + t * 8];
        pp[0] = __float_as_uint(w0 / s);
        pp[1] = __float_as_uint(w1 / s);
        pp[2] = __float_as_uint(w2 / s);
        pp[3] = __float_as_uint(bias[t]);
        int aa[3] = {a0, a1, a2};
        #pragma unroll
        for (int k = 0; k < 3; ++k) {
            int a = aa[k];
            int c = a / 9, r = a % 9, i = r / 3, j = r % 3;
            pp[4 + k] = (unsigned int)(c * (TH2 * TW2) + i * TW2 + j);
        }
        pp[7] = 0u;
    }

    wait_async_zero();      // drain this wave's async global->LDS copies
    __syncthreads();        // make tile + params visible workgroup-wide

    // ---- phase 2: 3-tap gather from LDS, one pixel per thread, 32 ch each ----
    const float*        tf = (const float*)smem;
    const float*        pf = (const float*)(smem + TILE_DW);
    const unsigned int* pu = smem + TILE_DW;

    const int p   = t & 63;          // pixel id in 8x8 tile
    const int g   = t >> 6;          // channel group 0..3
    const int ty  = p >> 3;
    const int tx  = p & 7;
    const int pix = ty * TW2 + tx;   // LDS pixel offset (bank-conflict-free per wave)
    const int h   = ty0 + ty;
    const int w   = tx0 + tx;

    float* ob = out + (size_t)b * COUT * (HW * HW) + (size_t)h * HW + w;
    for (int ch = g; ch < COUT; ch += 4) {
        const float*        q  = pf + ch * 8;   // wave-uniform -> LDS broadcast
        const unsigned int* qu = pu + ch * 8;
        float wn0 = q[0], wn1 = q[1], wn2 = q[2], bv = q[3];
        int   o0 = (int)qu[4], o1 = (int)qu[5], o2 = (int)qu[6];
        float acc = fmaf(wn0, tf[o0 + pix], bv);
        acc       = fmaf(wn1, tf[o1 + pix], acc);
        acc       = fmaf(wn2, tf[o2 + pix], acc);
        // streamed output, never re-read: nontemporal keeps x resident in L2
        __builtin_nontemporal_store(acc, ob + (size_t)ch * (HW * HW));
    }
}

extern "C" void kernel_launch(void* const* d_in, const int* in_sizes, int n_in,
                              void* d_out, int out_size, void* d_ws, size_t ws_size,
                              hipStream_t stream) {
    (void)in_sizes; (void)n_in; (void)out_size; (void)d_ws; (void)ws_size;
    const float* x       = (const float*)d_in[0];
    const float* weights = (const float*)d_in[1];
    const float* bias    = (const float*)d_in[2];
    const int*   conn    = (const int*)d_in[3];
    float*       out     = (float*)d_out;

    dim3 grid(TILES_X * TILES_Y * BATCH);   // 1568 workgroups
    dim3 block(256);                        // 8 wave32s
    uniconv_gather_kernel<<<grid, block, 0, stream>>>(x, weights, bias, conn, out);
}